// NoisyTopKRouter_16664473108707
// MI455X (gfx1250) — compile-verified
//
#include <hip/hip_runtime.h>
#include <stdint.h>
#include <string.h>

// Noisy top-2 router for MI455X (gfx1250, wave32).
// B x 64 f32 rows, bandwidth-bound (192 MiB @ 23.3 TB/s ~ 8.6 us floor).
// Data path: global -> LDS via GLOBAL_LOAD_ASYNC_TO_LDS_B128 (ASYNCcnt),
// per-thread top-2 scan from padded LDS rows, coalesced NT float2 stores.

#define EXPERTS        64
#define BLOCK_THREADS  64
#define ROWS_PER_BLOCK 64
#define ROW_WORDS      68                       // 64 data + 4 pad -> 272B stride, 16B aligned
#define NOISE_OFF_W    (ROWS_PER_BLOCK * ROW_WORDS)

// Low 32 bits of a flat pointer into LDS == the wave-relative LDS byte address
// (flat LDS aperture maps by truncation, ISA 10.2), so plain ptrtoint works.
__device__ __forceinline__ uint32_t lds_addr_of(const void* p) {
  return (uint32_t)(uintptr_t)p;
}

__global__ __launch_bounds__(BLOCK_THREADS)
void noisy_topk_router_kernel(const float* __restrict__ logits,
                              const float* __restrict__ noise,
                              float* __restrict__ out) {
  __shared__ float  s_buf[2 * ROWS_PER_BLOCK * ROW_WORDS]; // logits rows, then noise rows
  __shared__ float4 s_res[ROWS_PER_BLOCK];                 // (w1, w2, idx1, idx2) per row

  const int tid  = threadIdx.x;
  const int lane = tid & 31;
  const int wav  = tid >> 5;

  // ---------------- Phase 1: async copy global -> LDS (coalesced b128) -------------
  // Wave `wav` owns rows [wav*32, wav*32+32) of this block: an 8 KiB contiguous
  // region per input array. Iteration i copies 512B = rows 2i..2i+1 of that region.
  const uint64_t gLog =
      (uint64_t)(uintptr_t)(logits + (size_t)blockIdx.x * ROWS_PER_BLOCK * EXPERTS);
  const uint64_t gNoi =
      (uint64_t)(uintptr_t)(noise + (size_t)blockIdx.x * ROWS_PER_BLOCK * EXPERTS);

  uint32_t voff = (uint32_t)wav * 8192u + (uint32_t)lane * 16u; // byte off in wave region
  const int rr0 = wav * 32 + (lane >> 4);                       // first row this lane feeds
  uint32_t ldsLog = lds_addr_of(&s_buf[rr0 * ROW_WORDS]) + (uint32_t)(lane & 15) * 16u;
  uint32_t ldsNoi = ldsLog + (uint32_t)NOISE_OFF_W * 4u;

#pragma unroll
  for (int i = 0; i < 16; ++i) {
    // GVS mode: vdst = LDS byte address, vaddr = u32 offset, saddr = u64 base.
    asm volatile("global_load_async_to_lds_b128 %0, %1, %2"
                 :: "v"(ldsLog), "v"(voff), "s"(gLog) : "memory");
    asm volatile("global_load_async_to_lds_b128 %0, %1, %2"
                 :: "v"(ldsNoi), "v"(voff), "s"(gNoi) : "memory");
    ldsLog += 2u * ROW_WORDS * 4u; // advance two rows in LDS (544B)
    ldsNoi += 2u * ROW_WORDS * 4u;
    voff   += 512u;                // advance two rows in global
  }
  // Explicit asm wait with "memory" clobber: guarantees the compiler cannot hoist
  // the LDS reads of phase 2 above the async-copy completion point.
  asm volatile("s_wait_asynccnt 0x0" ::: "memory");

  // ---------------- Phase 2: per-thread exact top-2 scan of own row -----------------
  // Row stride 272B is 16B-aligned; b128 LDS reads, worst 2-way bank conflict.
  const float4* Lrow = (const float4*)&s_buf[tid * ROW_WORDS];
  const float4* Nrow = (const float4*)&s_buf[NOISE_OFF_W + tid * ROW_WORDS];

  float v1 = -__builtin_inff(), v2 = -__builtin_inff();
  int   i1 = 0, i2 = 0;
#pragma unroll
  for (int c = 0; c < 16; ++c) {
    const float4 lg = Lrow[c];
    const float4 nz = Nrow[c];
    const float vv[4] = {lg.x + nz.x, lg.y + nz.y, lg.z + nz.z, lg.w + nz.w};
#pragma unroll
    for (int u = 0; u < 4; ++u) {
      const float v = vv[u];
      const int   j = c * 4 + u;
      if (v > v1) {            // strict > keeps lower index on ties (jax top_k order)
        v2 = v1; i2 = i1; v1 = v; i1 = j;
      } else if (v > v2) {
        v2 = v; i2 = j;
      }
    }
  }
  // softmax over sorted [v1, v2], v1 >= v2 -> numerically safe
  const float e2  = __expf(v2 - v1);
  const float inv = 1.0f / (1.0f + e2);
  s_res[tid] = make_float4(inv, e2 * inv, __int_as_float(i1), __int_as_float(i2));

  __syncthreads();

  // ---------------- Phase 3: coalesced scatter, one row per wave-iteration ----------
  // Lane l emits output experts {2l, 2l+1} as one non-temporal b64 store (256B/row).
  float* outBlk = out + (size_t)blockIdx.x * ROWS_PER_BLOCK * EXPERTS;
  const int e0 = lane * 2;
  const int e1 = e0 + 1;
#pragma unroll 4
  for (int k = 0; k < 32; ++k) {
    const int    r  = wav * 32 + k;
    const float4 q  = s_res[r];                 // ds_load_b128 broadcast (conflict-free)
    const int    j1 = __float_as_int(q.z);
    const int    j2 = __float_as_int(q.w);
    const float  o0 = (e0 == j1) ? q.x : ((e0 == j2) ? q.y : 0.0f);
    const float  o1 = (e1 == j1) ? q.x : ((e1 == j2) ? q.y : 0.0f);
    float2 st = make_float2(o0, o1);
    double bits;
    __builtin_memcpy(&bits, &st, 8);
    __builtin_nontemporal_store(bits, (double*)(outBlk + (size_t)r * EXPERTS + e0));
  }
}

extern "C" void kernel_launch(void* const* d_in, const int* in_sizes, int n_in,
                              void* d_out, int out_size, void* d_ws, size_t ws_size,
                              hipStream_t stream) {
  (void)n_in; (void)out_size; (void)d_ws; (void)ws_size;
  const float* logits = (const float*)d_in[0];
  const float* noise  = (const float*)d_in[1];
  float*       out    = (float*)d_out;

  const int B      = in_sizes[0] / EXPERTS;     // 262144 in reference setup
  const int blocks = B / ROWS_PER_BLOCK;        // B is a multiple of 64
  noisy_topk_router_kernel<<<blocks, BLOCK_THREADS, 0, stream>>>(logits, noise, out);
}